// GeneralizedHausdorffDistance_28406913696125
// MI455X (gfx1250) — compile-verified
//
#include <hip/hip_runtime.h>
#include <math.h>

// ---------------------------------------------------------------------------
// Generalized Hausdorff Distance, CDNA5 (gfx1250)
//   d^2 via V_WMMA_F32_16X16X4_F32, transposed orientation:
//     A row g = [ny0, ny1, |ny|^2, 1]  (gt points, loop-invariant, 4 tiles)
//     B col p = [-2nx0, -2nx1, 1, |nx|^2]  (pixels, rebuilt per tile)
//   => lane holds a PIXEL column: term1 min over g is in-lane (1 shuffle/tile),
//      term2 min over pixels is a per-lane running min reduced once at the end.
//   clamp/min on loop-carried or shuffled values via raw inline-asm
//   v_max_num_f32 / v_min_num_f32 (dodges backend re-canonicalization of PHIs),
//   sqrt via raw v_sqrt_f32 (operands non-negative, non-denormal).
// ---------------------------------------------------------------------------

typedef float v2f __attribute__((ext_vector_type(2)));
typedef float v8f __attribute__((ext_vector_type(8)));

#define GHD_H 512
#define GHD_W 512
#define GHD_P (GHD_H * GHD_W)        // 262144 pixels
#define GHD_B 4
#define GHD_G 64                     // NGT
#define GHD_MAXD 724.0773439350246f  // sqrt(512^2+512^2)
#define GHD_EPS 1e-6f
#define GHD_INF __uint_as_float(0x7F800000u)

#define WAVES_PER_BLOCK 8
#define BLOCKS_PER_BATCH 32
#define NWAVES (WAVES_PER_BLOCK * BLOCKS_PER_BATCH) // 256 waves per batch
#define NTILES (GHD_P / 16)                         // 16384 pixel tiles per batch
#define TILES_PER_WAVE (NTILES / NWAVES)            // 64

// workspace layout (floats):
//  [0,               256)  : per-(batch,g) column minima (atomicMin on uint bits)
//  [256,            1280)  : per-(batch,wave) term1 partial sums
//  [1280,           2304)  : per-(batch,wave) sum(prob_map) partials
#define WS_MIN_OFF 0
#define WS_T1_OFF  (GHD_B * GHD_G)
#define WS_PM_OFF  (WS_T1_OFF + GHD_B * NWAVES)

// single-instruction max(x, 0): opaque to the backend's canonicalization
__device__ __forceinline__ float clamp0(float x) {
    float r;
    asm("v_max_num_f32 %0, 0, %1" : "=v"(r) : "v"(x));
    return r;
}
// single-instruction minimumNumber: opaque, avoids canonicalize of PHI/shuffle inputs
__device__ __forceinline__ float fmin_raw(float a, float b) {
    float r;
    asm("v_min_num_f32 %0, %1, %2" : "=v"(r) : "v"(a), "v"(b));
    return r;
}

__global__ void ghd_init(float* ws) {
    int i = blockIdx.x * blockDim.x + threadIdx.x;
    if (i < GHD_B * GHD_G) ws[WS_MIN_OFF + i] = GHD_INF;
}

__global__ __launch_bounds__(256)
void ghd_main(const float* __restrict__ prob_map,
              const float* __restrict__ prob_y,
              const float* __restrict__ gt,
              const int*   __restrict__ orig_sizes,
              float* __restrict__ ws) {
    const int b    = blockIdx.y;
    const int lane = threadIdx.x & 31;
    const int wv   = blockIdx.x * WAVES_PER_BLOCK + (threadIdx.x >> 5); // wave id in batch
    const int c    = lane & 15;   // column index (pixel within tile / gt within A tile)
    const int hi   = lane >> 4;   // half-wave select

    const float nf0 = (float)orig_sizes[b * 2 + 0] * (1.0f / 512.0f);
    const float nf1 = (float)orig_sizes[b * 2 + 1] * (1.0f / 512.0f);

    // --- A operands: 4 gt tiles, loop-invariant ---
    // A (16x4 f32): v0 = K0 (lanes<16, M=c) / K2 (lanes>=16, M=c); v1 = K1 / K3
    // row g = [ny0, ny1, |ny|^2, 1]
    v2f Aop[4];
#pragma unroll
    for (int t = 0; t < 4; ++t) {
        const int g     = t * 16 + c;
        const float ny0 = nf0 * gt[(b * GHD_G + g) * 2 + 0];
        const float ny1 = nf1 * gt[(b * GHD_G + g) * 2 + 1];
        const float sy  = ny0 * ny0 + ny1 * ny1;
        Aop[t].x = hi ? sy   : ny0;
        Aop[t].y = hi ? 1.0f : ny1;
    }

    // py for this half's D rows: D vgpr r of tile t holds gt g = t*16 + 8*hi + r
    float pyv[4][8];
#pragma unroll
    for (int t = 0; t < 4; ++t)
#pragma unroll
        for (int r = 0; r < 8; ++r)
            pyv[t][r] = prob_y[b * GHD_G + t * 16 + 8 * hi + r];

    const float* pm = prob_map + (size_t)b * GHD_P;

    float acc_t1 = 0.0f, acc_pm = 0.0f;
    float accmin2[4][8];
#pragma unroll
    for (int t = 0; t < 4; ++t)
#pragma unroll
        for (int r = 0; r < 8; ++r) accmin2[t][r] = GHD_INF;

    const float hmask = (hi == 0) ? 1.0f : 0.0f;  // select, keeps EXEC full for WMMA

    for (int it = 0; it < TILES_PER_WAVE; ++it) {
        const int tile = wv * TILES_PER_WAVE + it;
        const int p0   = tile * 16;

        // --- B operand: this tile's 16 pixel columns ---
        // B (4x16 f32): v0 = K0 (lanes<16, N=c) / K2 (lanes>=16, N=c); v1 = K1 / K3
        // col p = [-2nx0, -2nx1, 1, |nx|^2]
        const int   pl  = p0 + c;
        const float fi  = (float)(pl >> 9);      // row index (W = 512)
        const float fj  = (float)(pl & 511);     // col index
        const float nx0 = nf0 * fi;
        const float nx1 = nf1 * fj;
        const float sx  = nx0 * nx0 + nx1 * nx1;
        v2f Bop;
        Bop.x = hi ? 1.0f : (-2.0f * nx0);
        Bop.y = hi ? sx   : (-2.0f * nx1);

        // per-lane pixel prob (halves duplicate; counted once via hmask)
        const float p_c  = pm[pl];
        const float base = fmaf(-p_c, GHD_MAXD, GHD_MAXD);  // (1-p)*MAX_DIST

        float mloc = GHD_INF;  // min over this half's 32 g of py_g * d
#pragma unroll
        for (int t = 0; t < 4; ++t) {
            v8f c0 = {};
            // D = A_t(16x4 gt) x B(4x16 pixels) = d^2, rows = gt, cols = pixels
            v8f d2 = __builtin_amdgcn_wmma_f32_16x16x4_f32(
                false, Aop[t], false, Bop, (short)0, c0, false, false);

            float w[8];
#pragma unroll
            for (int r = 0; r < 8; ++r) {
                const float d = __builtin_amdgcn_sqrtf(clamp0(d2[r]));
                w[r] = pyv[t][r] * d;                               // py_g * d
                // (1-p)*MAX + p*py*d ; raw min: no PHI re-canonicalization
                accmin2[t][r] = fmin_raw(accmin2[t][r], fmaf(p_c, w[r], base));
            }
            // min tree over the 8 rows of this tile (mul results: canonical, dual-issuable)
            const float m0 = fminf(fminf(w[0], w[1]), fminf(w[2], w[3]));
            const float m1 = fminf(fminf(w[4], w[5]), fminf(w[6], w[7]));
            mloc = fmin_raw(mloc, fminf(m0, m1));
        }
        // merge g%16 in 0..7 half with g%16 in 8..15 half: full min over all 64 gt points
        mloc = fmin_raw(mloc, __shfl_xor(mloc, 16, 32));

        // term1 += p_x * min_g(py*d); count each pixel once (hi==0 half only)
        acc_t1 = fmaf(hmask * p_c, mloc, acc_t1);
        acc_pm = fmaf(hmask, p_c, acc_pm);
    }

    // --- epilogue: term1/pm sums across the 16 columns (hi==0 lanes hold data) ---
    acc_t1 += __shfl_xor(acc_t1, 1, 32);
    acc_t1 += __shfl_xor(acc_t1, 2, 32);
    acc_t1 += __shfl_xor(acc_t1, 4, 32);
    acc_t1 += __shfl_xor(acc_t1, 8, 32);
    acc_pm += __shfl_xor(acc_pm, 1, 32);
    acc_pm += __shfl_xor(acc_pm, 2, 32);
    acc_pm += __shfl_xor(acc_pm, 4, 32);
    acc_pm += __shfl_xor(acc_pm, 8, 32);
    if (lane == 0) {
        ws[WS_T1_OFF + b * NWAVES + wv] = acc_t1;
        ws[WS_PM_OFF + b * NWAVES + wv] = acc_pm;
    }

    // --- epilogue: term2 minima across the 16 pixel columns of each half ---
#pragma unroll
    for (int t = 0; t < 4; ++t)
#pragma unroll
        for (int r = 0; r < 8; ++r) {
            float m = accmin2[t][r];
            m = fmin_raw(m, __shfl_xor(m, 1, 32));
            m = fmin_raw(m, __shfl_xor(m, 2, 32));
            m = fmin_raw(m, __shfl_xor(m, 4, 32));
            m = fmin_raw(m, __shfl_xor(m, 8, 32));
            accmin2[t][r] = m;
        }
    if (c == 0) {  // lanes 0 and 16: each half owns its own 32 gt rows
        unsigned int* wsm = (unsigned int*)(ws + WS_MIN_OFF);
#pragma unroll
        for (int t = 0; t < 4; ++t)
#pragma unroll
            for (int r = 0; r < 8; ++r)
                atomicMin(&wsm[b * GHD_G + t * 16 + 8 * hi + r],
                          __float_as_uint(accmin2[t][r]));
    }
}

__global__ void ghd_final(const float* __restrict__ prob_y,
                          const float* __restrict__ ws,
                          float* __restrict__ out) {
    if (threadIdx.x != 0 || blockIdx.x != 0) return;
    float t1_sum = 0.0f, t2_sum = 0.0f;
    for (int b = 0; b < GHD_B; ++b) {
        float s1 = 0.0f, sp = 0.0f;
        for (int w = 0; w < NWAVES; ++w) {
            s1 += ws[WS_T1_OFF + b * NWAVES + w];
            sp += ws[WS_PM_OFF + b * NWAVES + w];
        }
        float smin = 0.0f, spy = 0.0f;
        for (int g = 0; g < GHD_G; ++g) {
            smin += ws[WS_MIN_OFF + b * GHD_G + g];
            const float py = prob_y[b * GHD_G + g];
            spy += (py > 0.2f) ? py : 0.0f;
        }
        t1_sum += s1 / (sp + GHD_EPS);
        t2_sum += smin / (spy + GHD_EPS);
    }
    out[0] = (t1_sum + t2_sum) * (1.0f / (float)GHD_B);
}

extern "C" void kernel_launch(void* const* d_in, const int* in_sizes, int n_in,
                              void* d_out, int out_size, void* d_ws, size_t ws_size,
                              hipStream_t stream) {
    const float* prob_map   = (const float*)d_in[0];
    const float* prob_y     = (const float*)d_in[1];
    const float* gt         = (const float*)d_in[2];
    const int*   orig_sizes = (const int*)d_in[3];
    float* out = (float*)d_out;
    float* ws  = (float*)d_ws;

    ghd_init<<<1, 256, 0, stream>>>(ws);
    dim3 grid(BLOCKS_PER_BATCH, GHD_B);
    ghd_main<<<grid, 256, 0, stream>>>(prob_map, prob_y, gt, orig_sizes, ws);
    ghd_final<<<1, 32, 0, stream>>>(prob_y, ws, out);
}